// GatedDeltaRule_91079076479233
// MI455X (gfx1250) — compile-verified
//
#include <hip/hip_runtime.h>
#include <math.h>

#define DIMK 1024
#define CC   128
#define BB   8
#define TT   2048
#define MTOT (BB*TT)   // 16384

typedef __attribute__((ext_vector_type(2))) float v2f;
typedef __attribute__((ext_vector_type(8))) float v8f;

// ---------------------------------------------------------------------------
// fp32 WMMA GEMM: Y[M,N] = X[M,K] @ W[K,N] + bias[N]
// Block tile 64x64, BK=32, 256 threads = 8 waves, wave tile 16x32.
// Software pipelined float4 staging, ping-pong LDS, 1 barrier per K-block,
// W transposed in LDS so B fragments are contiguous ds_load_b64.
// ---------------------------------------------------------------------------
__global__ __launch_bounds__(256)
void gemm_f32_wmma(const float* __restrict__ X, const float* __restrict__ W,
                   const float* __restrict__ bias, float* __restrict__ Y,
                   int M, int N, int K)
{
    __shared__ float Xs[2][64 * 36];
    __shared__ float Ws[2][64 * 34];   // transposed [n][k]

    const int tid  = threadIdx.x;
    const int wv   = tid >> 5;
    const int lane = tid & 31;
    const int m0   = blockIdx.x * 64;
    const int n0   = blockIdx.y * 64;
    const int mw   = (wv >> 1) * 16;
    const int nw   = (wv & 1) * 32;
    const int half = lane >> 4;
    const int l16  = lane & 15;

    const int xr  = tid >> 3;
    const int xc4 = (tid & 7) * 4;
    const int wk  = tid >> 4;
    const int wn4 = (tid & 15) * 4;

    const float* px0 = X + (size_t)(m0 + xr) * K + xc4;
    const float* px1 = px0 + (size_t)32 * K;
    const float* pw0 = W + (size_t)wk * N + n0 + wn4;
    const float* pw1 = pw0 + (size_t)16 * N;

    const int nk = K >> 5;

    float4 x0 = *(const float4*)px0;  px0 += 32;
    float4 x1 = *(const float4*)px1;  px1 += 32;
    float4 w0 = *(const float4*)pw0;  pw0 += (size_t)32 * N;
    float4 w1 = *(const float4*)pw1;  pw1 += (size_t)32 * N;
    {
        *(float4*)&Xs[0][xr * 36 + xc4]        = x0;
        *(float4*)&Xs[0][(xr + 32) * 36 + xc4] = x1;
        #pragma unroll
        for (int j = 0; j < 4; ++j) Ws[0][(wn4 + j) * 34 + wk]      = (&w0.x)[j];
        #pragma unroll
        for (int j = 0; j < 4; ++j) Ws[0][(wn4 + j) * 34 + wk + 16] = (&w1.x)[j];
    }

    v8f acc0 = {};
    v8f acc1 = {};

    for (int i = 0; i < nk; ++i) {
        const int p = i & 1;
        __syncthreads();

        v2f af[8], bf0[8], bf1[8];
        const float* xb  = &Xs[p][(mw + l16) * 36 + 2 * half];
        const float* w0b = &Ws[p][(nw + l16) * 34 + 2 * half];
        const float* w1b = &Ws[p][(nw + 16 + l16) * 34 + 2 * half];
        #pragma unroll
        for (int ks = 0; ks < 8; ++ks) {
            af[ks]  = *(const v2f*)(xb  + ks * 4);
            bf0[ks] = *(const v2f*)(w0b + ks * 4);
            bf1[ks] = *(const v2f*)(w1b + ks * 4);
        }

        const bool more = (i + 1) < nk;
        if (more) {
            x0 = *(const float4*)px0;  px0 += 32;
            x1 = *(const float4*)px1;  px1 += 32;
            w0 = *(const float4*)pw0;  pw0 += (size_t)32 * N;
            w1 = *(const float4*)pw1;  pw1 += (size_t)32 * N;
        }

        #pragma unroll
        for (int ks = 0; ks < 8; ++ks) {
            acc0 = __builtin_amdgcn_wmma_f32_16x16x4_f32(
                       false, af[ks], false, bf0[ks], (short)0, acc0, false, false);
            acc1 = __builtin_amdgcn_wmma_f32_16x16x4_f32(
                       false, af[ks], false, bf1[ks], (short)0, acc1, false, false);
        }

        if (more) {
            const int np = p ^ 1;
            *(float4*)&Xs[np][xr * 36 + xc4]        = x0;
            *(float4*)&Xs[np][(xr + 32) * 36 + xc4] = x1;
            #pragma unroll
            for (int j = 0; j < 4; ++j) Ws[np][(wn4 + j) * 34 + wk]      = (&w0.x)[j];
            #pragma unroll
            for (int j = 0; j < 4; ++j) Ws[np][(wn4 + j) * 34 + wk + 16] = (&w1.x)[j];
        }
    }

    #pragma unroll
    for (int r = 0; r < 8; ++r) {
        int m = m0 + mw + r + 8 * half;
        int n = n0 + nw + l16;
        Y[(size_t)m * N + n]      = acc0[r] + bias[n];
        Y[(size_t)m * N + n + 16] = acc1[r] + bias[n + 16];
    }
}

// ---------------------------------------------------------------------------
// alpha/beta: sigmoid(x @ Wa + ba), sigmoid(x @ Wb + bb). One wave per row.
// ---------------------------------------------------------------------------
__global__ __launch_bounds__(256)
void ab_kernel(const float* __restrict__ x,
               const float* __restrict__ Wa, const float* __restrict__ ba,
               const float* __restrict__ Wb, const float* __restrict__ bb,
               float* __restrict__ aout, float* __restrict__ bout)
{
    __shared__ float wa[DIMK];
    __shared__ float wb[DIMK];
    const int tid = threadIdx.x;
    for (int i = tid; i < DIMK; i += 256) { wa[i] = Wa[i]; wb[i] = Wb[i]; }
    __syncthreads();

    const int wv = tid >> 5, lane = tid & 31;
    const int row = blockIdx.x * 8 + wv;
    const float* xr = x + (size_t)row * DIMK;
    float sa = 0.f, sb = 0.f;
    #pragma unroll 4
    for (int j = lane; j < DIMK; j += 32) {
        float xv = xr[j];
        sa = fmaf(xv, wa[j], sa);
        sb = fmaf(xv, wb[j], sb);
    }
    #pragma unroll
    for (int off = 16; off; off >>= 1) {
        sa += __shfl_xor(sa, off, 32);
        sb += __shfl_xor(sb, off, 32);
    }
    if (lane == 0) {
        aout[row] = 1.f / (1.f + __expf(-(sa + ba[0])));
        bout[row] = 1.f / (1.f + __expf(-(sb + bb[0])));
    }
}

// ---------------------------------------------------------------------------
// k <- k / max(||k||, 1e-12) per row; g <- sigmoid(g). One wave per row.
// ---------------------------------------------------------------------------
__global__ __launch_bounds__(512)
void norm_sig_kernel(float* __restrict__ k, float* __restrict__ g)
{
    const int tid = threadIdx.x;
    const int wv = tid >> 5, lane = tid & 31;
    const int row = blockIdx.x * 16 + wv;
    float* kr = k + (size_t)row * CC;
    float v0 = kr[lane], v1 = kr[lane + 32], v2 = kr[lane + 64], v3 = kr[lane + 96];
    float ss = v0 * v0 + v1 * v1 + v2 * v2 + v3 * v3;
    #pragma unroll
    for (int off = 16; off; off >>= 1) ss += __shfl_xor(ss, off, 32);
    float inv = 1.f / fmaxf(sqrtf(ss), 1e-12f);
    kr[lane]      = v0 * inv;
    kr[lane + 32] = v1 * inv;
    kr[lane + 64] = v2 * inv;
    kr[lane + 96] = v3 * inv;

    float* gr = g + (size_t)row * CC;
    #pragma unroll
    for (int j = 0; j < 4; ++j) {
        float gv = gr[lane + 32 * j];
        gr[lane + 32 * j] = 1.f / (1.f + __expf(-gv));
    }
}

// ---------------------------------------------------------------------------
// Barrier-free scan: 128 fully independent waves (8 batches x 16 row-groups),
// 32 blocks x 128 threads -> one wave per SIMD across 32 WGPs.
// Wave w owns S rows [8w,8w+8); lane owns cols 4*lane..4*lane+3 (32 VGPRs).
// Per step: k_t as one float4/lane, q/v rows as wave-uniform float4 loads,
// Sk via shuffle tree, rank-1 register update, partial o streamed to global.
// No LDS, no barriers, no atomics; 1-step software pipeline + prefetch.
// ---------------------------------------------------------------------------
__global__ __launch_bounds__(128)
void scan_nosync_kernel(const float* __restrict__ q, const float* __restrict__ k,
                        const float* __restrict__ v,
                        const float* __restrict__ ag, const float* __restrict__ bg,
                        float* __restrict__ part)
{
    const int tid  = threadIdx.x;
    const int wv   = tid >> 5;
    const int lane = tid & 31;
    const int widx = blockIdx.x * 4 + wv;          // 0..127
    const int b    = widx >> 4;
    const int wgrp = widx & 15;
    const int row0 = __builtin_amdgcn_readfirstlane(wgrp * 8);
    const size_t base = (size_t)b * TT * CC;

    const float* kp = k + base + 4 * lane;
    const float* qp = q + base + row0;
    const float* vp = v + base + row0;
    const float* pa = ag + (size_t)b * TT;
    const float* pb = bg + (size_t)b * TT;
    float* pp = part + ((size_t)b * TT * 16 + wgrp) * CC + 4 * lane;

    float S[8][4];
    #pragma unroll
    for (int r = 0; r < 8; ++r)
        #pragma unroll
        for (int j = 0; j < 4; ++j) S[r][j] = 0.f;

    // prologue: loads for t = 0
    float4 kc  = *(const float4*)kp;
    float4 q0c = *(const float4*)qp;
    float4 q1c = *(const float4*)(qp + 4);
    float4 v0c = *(const float4*)vp;
    float4 v1c = *(const float4*)(vp + 4);
    float  a_s = pa[0];
    float  b_s = pb[0];

    for (int t = 0; t < TT; ++t) {
        // issue next step's loads (in flight during this step's math)
        float4 kn = {}, q0n = {}, q1n = {}, v0n = {}, v1n = {};
        float an = 0.f, bn = 0.f;
        if (t + 1 < TT) {
            const size_t no = (size_t)(t + 1) * CC;
            kn  = *(const float4*)(kp + no);
            q0n = *(const float4*)(qp + no);
            q1n = *(const float4*)(qp + no + 4);
            v0n = *(const float4*)(vp + no);
            v1n = *(const float4*)(vp + no + 4);
            an  = pa[t + 1];
            bn  = pb[t + 1];
        }
        if (t + 4 < TT) {
            __builtin_prefetch(kp + (size_t)(t + 4) * CC, 0, 3);
            __builtin_prefetch(qp + (size_t)(t + 4) * CC, 0, 3);
            __builtin_prefetch(vp + (size_t)(t + 4) * CC, 0, 3);
        }

        const float kcv[4] = { kc.x, kc.y, kc.z, kc.w };
        const float qi[8]  = { q0c.x, q0c.y, q0c.z, q0c.w, q1c.x, q1c.y, q1c.z, q1c.w };
        const float vi[8]  = { v0c.x, v0c.y, v0c.z, v0c.w, v1c.x, v1c.y, v1c.z, v1c.w };

        // Sk_i = sum_j S[i][j]*k[j] (wave shuffle tree); u_i = b*v_i - a*b*Sk_i
        float u[8];
        #pragma unroll
        for (int r = 0; r < 8; ++r) {
            float pr = S[r][0] * kcv[0] + S[r][1] * kcv[1]
                     + S[r][2] * kcv[2] + S[r][3] * kcv[3];
            pr += __shfl_xor(pr, 16, 32);
            pr += __shfl_xor(pr,  8, 32);
            pr += __shfl_xor(pr,  4, 32);
            pr += __shfl_xor(pr,  2, 32);
            pr += __shfl_xor(pr,  1, 32);
            u[r] = b_s * vi[r] - a_s * b_s * pr;
        }

        // S = a*S + u (x) k ; partial o_j = sum_{i in rows} q_i * S_new[i][j]
        float po[4] = {0.f, 0.f, 0.f, 0.f};
        #pragma unroll
        for (int r = 0; r < 8; ++r) {
            #pragma unroll
            for (int j = 0; j < 4; ++j) {
                float s = fmaf(a_s, S[r][j], u[r] * kcv[j]);
                S[r][j] = s;
                po[j] = fmaf(qi[r], s, po[j]);
            }
        }
        float4 po4 = { po[0], po[1], po[2], po[3] };
        *(float4*)(pp + (size_t)t * 16 * CC) = po4;   // fire-and-forget store

        kc = kn; q0c = q0n; q1c = q1n; v0c = v0n; v1c = v1n;
        a_s = an; b_s = bn;
    }
}

// ---------------------------------------------------------------------------
// ored[m][c] = (sum_w part[m][w][c]) * g[m][c]; one float4 per thread.
// ---------------------------------------------------------------------------
__global__ __launch_bounds__(256)
void reduce_gate_kernel(const float* __restrict__ part, const float* __restrict__ g,
                        float* __restrict__ ored)
{
    const int gid = blockIdx.x * 256 + threadIdx.x;
    const int m  = gid >> 5;
    const int c4 = (gid & 31) * 4;
    const float* p = part + (size_t)m * 16 * CC + c4;
    float4 s = {0.f, 0.f, 0.f, 0.f};
    #pragma unroll
    for (int w = 0; w < 16; ++w) {
        float4 t = *(const float4*)(p + w * CC);
        s.x += t.x; s.y += t.y; s.z += t.z; s.w += t.w;
    }
    float4 gv = *(const float4*)(g + (size_t)m * CC + c4);
    s.x *= gv.x; s.y *= gv.y; s.z *= gv.z; s.w *= gv.w;
    *(float4*)(ored + (size_t)m * CC + c4) = s;
}

// ---------------------------------------------------------------------------
// Fallback scan (LDS-reduction, 2 barriers/step) if workspace is small.
// ---------------------------------------------------------------------------
__global__ __launch_bounds__(512)
void scan_kernel(const float* __restrict__ q, const float* __restrict__ k,
                 const float* __restrict__ v, const float* __restrict__ g,
                 const float* __restrict__ ag, const float* __restrict__ bg,
                 float* __restrict__ o)
{
    __shared__ float inbuf[2][520];
    __shared__ float obuf[2][128];

    const int tid  = threadIdx.x;
    const int wv   = tid >> 5;
    const int lane = tid & 31;
    const int b    = blockIdx.x;
    const size_t base = (size_t)b * TT * CC;
    const int row0 = wv * 8;

    const int lidx = tid & 127;
    const float* srcp;
    if      (tid < 128) srcp = k + base + lidx;
    else if (tid < 256) srcp = q + base + lidx;
    else if (tid < 384) srcp = v + base + lidx;
    else                srcp = g + base + lidx;

    const float* pa = ag + (size_t)b * TT;
    const float* pb = bg + (size_t)b * TT;
    float* op = o + base + tid;

    float S[8][4];
    #pragma unroll
    for (int r = 0; r < 8; ++r)
        #pragma unroll
        for (int j = 0; j < 4; ++j) S[r][j] = 0.f;

    if (tid < 128) { obuf[0][tid] = 0.f; obuf[1][tid] = 0.f; }

    for (int t = 0; t < TT; ++t) {
        const int p = t & 1;
        inbuf[p][tid] = *srcp;
        if (t + 8 < TT) __builtin_prefetch(srcp + 8 * CC, 0, 3);
        srcp += CC;
        if (tid == 0) inbuf[p][512] = pa[t];
        if (tid == 1) inbuf[p][513] = pb[t];
        __syncthreads();

        const float a_s = inbuf[p][512];
        const float b_s = inbuf[p][513];
        float kc[4];
        #pragma unroll
        for (int j = 0; j < 4; ++j) kc[j] = inbuf[p][lane + 32 * j];

        float u[8];
        #pragma unroll
        for (int r = 0; r < 8; ++r) {
            float pr = S[r][0] * kc[0] + S[r][1] * kc[1]
                     + S[r][2] * kc[2] + S[r][3] * kc[3];
            pr += __shfl_xor(pr, 16, 32);
            pr += __shfl_xor(pr,  8, 32);
            pr += __shfl_xor(pr,  4, 32);
            pr += __shfl_xor(pr,  2, 32);
            pr += __shfl_xor(pr,  1, 32);
            float vix = inbuf[p][256 + row0 + r];
            u[r] = b_s * vix - a_s * b_s * pr;
        }

        float po[4] = {0.f, 0.f, 0.f, 0.f};
        #pragma unroll
        for (int r = 0; r < 8; ++r) {
            float qi = inbuf[p][128 + row0 + r];
            #pragma unroll
            for (int j = 0; j < 4; ++j) {
                float s = fmaf(a_s, S[r][j], u[r] * kc[j]);
                S[r][j] = s;
                po[j] = fmaf(qi, s, po[j]);
            }
        }
        #pragma unroll
        for (int j = 0; j < 4; ++j)
            atomicAdd(&obuf[p][lane + 32 * j], po[j]);
        __syncthreads();

        if (tid < 128) {
            *op = obuf[p][tid] * inbuf[p][384 + tid];
            op += CC;
            obuf[p][tid] = 0.f;
        }
    }
}

// ---------------------------------------------------------------------------
extern "C" void kernel_launch(void* const* d_in, const int* in_sizes, int n_in,
                              void* d_out, int out_size, void* d_ws, size_t ws_size,
                              hipStream_t stream)
{
    const float* x   = (const float*)d_in[0];
    const float* Wq  = (const float*)d_in[1];
    const float* bq  = (const float*)d_in[2];
    const float* Wk  = (const float*)d_in[3];
    const float* bk  = (const float*)d_in[4];
    const float* Wv  = (const float*)d_in[5];
    const float* bv  = (const float*)d_in[6];
    const float* Wa  = (const float*)d_in[7];
    const float* ba  = (const float*)d_in[8];
    const float* Wb  = (const float*)d_in[9];
    const float* bb  = (const float*)d_in[10];
    const float* Wg  = (const float*)d_in[11];
    const float* bg_ = (const float*)d_in[12];
    const float* Wo  = (const float*)d_in[13];
    const float* bo  = (const float*)d_in[14];
    float* out = (float*)d_out;

    const size_t NC = (size_t)MTOT * CC;
    float* ws  = (float*)d_ws;
    float* qb  = ws;
    float* kb  = qb + NC;
    float* vb  = kb + NC;
    float* gb  = vb + NC;
    float* ob  = gb + NC;
    float* av  = ob + NC;
    float* bv_ = av + MTOT;
    float* part = bv_ + MTOT;              // 16*NC floats (fast path only)

    const size_t need_fast = (5 * NC + 2 * (size_t)MTOT + 16 * NC) * sizeof(float);

    dim3 blk(256);
    dim3 gproj(MTOT / 64, CC / 64);
    gemm_f32_wmma<<<gproj, blk, 0, stream>>>(x, Wq, bq, qb, MTOT, CC, DIMK);
    gemm_f32_wmma<<<gproj, blk, 0, stream>>>(x, Wk, bk, kb, MTOT, CC, DIMK);
    gemm_f32_wmma<<<gproj, blk, 0, stream>>>(x, Wv, bv, vb, MTOT, CC, DIMK);
    gemm_f32_wmma<<<gproj, blk, 0, stream>>>(x, Wg, bg_, gb, MTOT, CC, DIMK);

    ab_kernel<<<MTOT / 8, 256, 0, stream>>>(x, Wa, ba, Wb, bb, av, bv_);
    norm_sig_kernel<<<MTOT / 16, 512, 0, stream>>>(kb, gb);

    if (ws_size >= need_fast) {
        scan_nosync_kernel<<<32, 128, 0, stream>>>(qb, kb, vb, av, bv_, part);
        reduce_gate_kernel<<<(MTOT * 32) / 256, 256, 0, stream>>>(part, gb, ob);
    } else {
        scan_kernel<<<BB, 512, 0, stream>>>(qb, kb, vb, gb, av, bv_, ob);
    }

    dim3 gout(MTOT / 64, DIMK / 64);
    gemm_f32_wmma<<<gout, blk, 0, stream>>>(ob, Wo, bo, out, MTOT, DIMK, CC);
}